// FEDformer_36885179138801
// MI455X (gfx1250) — compile-verified
//
#include <hip/hip_runtime.h>
#include <math.h>

// FEDformer block collapses (all ops linear in x) to:
//   out[b] = c0 + dot(flatten(x[b]) /*144 f32*/, M /*144 f32*/)
// Phase 1: tiny precompute of M[144] and c0 into d_ws.
// Phase 2: [65536 x 144] x [144] mat-vec via V_WMMA_F32_16X16X4_F32,
//          16 batches per wave, 36 chained K=4 WMMA steps (144 = 36*4).

typedef __attribute__((ext_vector_type(2))) float v2f;
typedef __attribute__((ext_vector_type(8))) float v8f;

#define Bsz   65536
#define Lseq  24
#define CIN   6
#define Dm    64
#define MODES 12
#define KTOT  (Lseq * CIN)   // 144

// ---------------- Phase 1: fold weights into M[144] + c0 -------------------
__global__ __launch_bounds__(256)
void fed_precompute(const float* __restrict__ W_emb,  // [6,64]
                    const float* __restrict__ b_emb,  // [64]
                    const float* __restrict__ pos,    // [1,24,64]
                    const float* __restrict__ w1,     // [64,64,12,2]
                    const float* __restrict__ W_dec,  // [64,1]
                    const float* __restrict__ b_dec,  // [1]
                    float* __restrict__ ws)           // out: M[144], c0 at [144]
{
    __shared__ float u_re[Dm][MODES];
    __shared__ float u_im[Dm][MODES];
    __shared__ float g[Lseq][Dm];
    __shared__ float red[256];
    const int tid = threadIdx.x;
    const float PI = 3.14159265358979323846f;

    // u[c,k] = a_k * sum_d (w1[c,d,k,0] + i*w1[c,d,k,1]) * W_dec[d]
    // a_0 = 1/24 (irfft DC, imag ignored), a_k = (2/24) e^{-i 2pi k/24} (t = L-1)
    for (int idx = tid; idx < Dm * MODES; idx += 256) {
        int c = idx / MODES, k = idx % MODES;
        float vre = 0.f, vim = 0.f;
        const float* wp = w1 + ((size_t)c * Dm * MODES + k) * 2;
        for (int d = 0; d < Dm; ++d) {
            float wd = W_dec[d];
            vre += wp[(size_t)d * (MODES * 2)]     * wd;
            vim += wp[(size_t)d * (MODES * 2) + 1] * wd;
        }
        float are, aim;
        if (k == 0) { are = 1.0f / 24.0f; aim = 0.0f; }
        else {
            float ph = 2.0f * PI * (float)k / 24.0f;
            are =  (2.0f / 24.0f) * cosf(ph);
            aim = -(2.0f / 24.0f) * sinf(ph);
        }
        u_re[c][k] = are * vre - aim * vim;
        u_im[c][k] = are * vim + aim * vre;
    }
    __syncthreads();

    // g[t,d] = sum_k Re(e^{-i 2pi k t/24} * u[d,k])
    for (int idx = tid; idx < Lseq * Dm; idx += 256) {
        int t = idx / Dm, d = idx % Dm;
        float s = 0.f;
        for (int k = 0; k < MODES; ++k) {
            float ph = 2.0f * PI * (float)(k * t) / 24.0f;
            s += u_re[d][k] * cosf(ph) + u_im[d][k] * sinf(ph);
        }
        g[t][d] = s;
    }
    __syncthreads();

    // M[t,ci] = sum_d W_emb[ci,d] * g[t,d]   (144 entries)
    if (tid < KTOT) {
        int t = tid / CIN, ci = tid % CIN;
        float s = 0.f;
        for (int d = 0; d < Dm; ++d) s += W_emb[ci * Dm + d] * g[t][d];
        ws[tid] = s;
    }

    // c0 = b_dec + sum_{t,d} (b_emb[d] + pos[t,d]) * g[t,d]
    float part = 0.f;
    for (int idx = tid; idx < Lseq * Dm; idx += 256) {
        int t = idx / Dm, d = idx % Dm;
        part += (b_emb[d] + pos[t * Dm + d]) * g[t][d];
    }
    red[tid] = part;
    __syncthreads();
    for (int ofs = 128; ofs > 0; ofs >>= 1) {
        if (tid < ofs) red[tid] += red[tid + ofs];
        __syncthreads();
    }
    if (tid == 0) ws[KTOT] = red[0] + b_dec[0];
}

// ---------------- Phase 2: WMMA mat-vec ------------------------------------
// Per wave: A tile = 16 batch rows x K=4 of flattened x.
// f32 16x16x4 A layout (ISA 7.12.2): lanes 0-15 -> M=lane, VGPR0/1 = K0/K1;
// lanes 16-31 -> M=lane-16, VGPR0/1 = K2/K3.  B (4x16) uses the same
// (vgpr, lane-half) -> K placement; we replicate M across all 16 columns so
// every D column holds the identical dot product (robust to N mapping).
__global__ __launch_bounds__(256)
void fed_wmma_dot(const float* __restrict__ x,   // [65536,24,6] = [65536,144]
                  const float* __restrict__ ws,  // M[144], c0
                  float* __restrict__ out)       // [65536]
{
    __shared__ __align__(16) float Ms[KTOT];
    const int tid = threadIdx.x;
    if (tid < KTOT) Ms[tid] = ws[tid];
    __syncthreads();

    const float c0   = ws[KTOT];
    const int lane   = tid & 31;
    const int wave   = tid >> 5;
    const int tile   = blockIdx.x * 128 + wave * 16;   // 8 waves * 16 rows
    const int row    = lane & 15;
    const int koff   = ((lane >> 4) & 1) * 2;          // 0 for K{0,1}, 2 for K{2,3}

    const float* xrow = x + (size_t)(tile + row) * KTOT + koff;
    const float* mrow = Ms + koff;

    v8f acc = {};
#pragma unroll
    for (int s = 0; s < KTOT / 4; ++s) {               // 36 WMMA steps
        v2f a = *(const v2f*)(xrow + 4 * s);           // global_load_b64
        v2f b = *(const v2f*)(mrow + 4 * s);           // ds_load_b64
        acc = __builtin_amdgcn_wmma_f32_16x16x4_f32(
            /*neg_a=*/false, a, /*neg_b=*/false, b,
            /*c_mod=*/(short)0, acc, /*reuse_a=*/false, /*reuse_b=*/false);
    }

    // D layout: VGPR j -> M=j (lanes 0-15), M=8+j (lanes 16-31); all columns equal.
    if (lane == 0) {
#pragma unroll
        for (int m = 0; m < 8; ++m) out[tile + m] = acc[m] + c0;
    } else if (lane == 16) {
#pragma unroll
        for (int m = 0; m < 8; ++m) out[tile + 8 + m] = acc[m] + c0;
    }
}

extern "C" void kernel_launch(void* const* d_in, const int* in_sizes, int n_in,
                              void* d_out, int out_size, void* d_ws, size_t ws_size,
                              hipStream_t stream) {
    const float* x     = (const float*)d_in[0];
    const float* W_emb = (const float*)d_in[1];
    const float* b_emb = (const float*)d_in[2];
    const float* pos   = (const float*)d_in[3];
    const float* w1    = (const float*)d_in[4];
    const float* W_dec = (const float*)d_in[5];
    const float* b_dec = (const float*)d_in[6];
    float* out = (float*)d_out;
    float* ws  = (float*)d_ws;

    fed_precompute<<<1, 256, 0, stream>>>(W_emb, b_emb, pos, w1, W_dec, b_dec, ws);
    // 65536 batches / (8 waves * 16 rows) = 512 blocks
    fed_wmma_dot<<<512, 256, 0, stream>>>(x, ws, out);
}